// SpikeFP32Floor_43860206027309
// MI455X (gfx1250) — compile-verified
//
#include <hip/hip_runtime.h>
#include <stdint.h>

// SpikeFP32Floor for MI455X (gfx1250).
// Memory-bound: 512 MB stream -> ~22us floor @ 23.3 TB/s. WMMA is inapplicable
// (bit-pack needs exact 32-bit integer accumulation). Optimization target is
// the CDNA5 async global->LDS data path + coalesced b128 traffic, wave32-local
// tiles (no barriers), and NT stores.

#define THREADS        256
#define VALS_PER_BLOCK 256
#define ROW_DW         36   // 32 data dwords + 4 pad: rows stay 16B-aligned,
                            // lane stride 36 dw -> <=2-way LDS bank conflict

#if defined(__has_builtin)
#if __has_builtin(__builtin_amdgcn_global_load_async_to_lds_b128) && \
    __has_builtin(__builtin_amdgcn_s_wait_asynccnt)
#define HAVE_ASYNC_LDS 1
#endif
#endif
#ifndef HAVE_ASYNC_LDS
#define HAVE_ASYNC_LDS 0
#endif

typedef float    f32x4 __attribute__((ext_vector_type(4)));
typedef uint32_t u32x4 __attribute__((ext_vector_type(4)));
typedef int      i32x4 __attribute__((ext_vector_type(4)));

// Address-space-qualified int4 types matching the async-LDS builtin signature:
// param0 is int4 addrspace(1)* (printed as "__device__" by clang), param1 is
// the LDS-side int4 addrspace(3)*.
typedef __attribute__((address_space(1))) i32x4 gbl_i32x4;
typedef __attribute__((address_space(3))) i32x4 lds_i32x4;

// floor() on the raw fp32 bit pattern, exactly mirroring the reference's
// select priority: mid-range trunc(-1) -> (e<127) override -> (e>=150) override.
__device__ __forceinline__ uint32_t floor_bits(uint32_t u) {
  const uint32_t s  = u >> 31;
  const uint32_t e  = (u >> 23) & 0xFFu;
  const uint32_t sh = 150u - e;  // #fractional mantissa bits when 127<=e<150
  const uint32_t fmask = (sh < 32u) ? ((1u << sh) - 1u) : 0xFFFFFFFFu;
  const uint32_t trunc = u & ~fmask;
  const uint32_t tm1 = __float_as_uint(__uint_as_float(trunc) - 1.0f);
  uint32_t res = ((s != 0u) && ((u & fmask) != 0u)) ? tm1 : trunc;
  res = (e < 127u)  ? (s ? 0xBF800000u : 0x00000000u) : res;  // |v|<1 cases
  res = (e >= 150u) ? u : res;                                // integer / inf / nan
  return res;
}

__device__ __forceinline__ void unpack_store(uint32_t res, float* gout) {
  #pragma unroll
  for (int k = 0; k < 8; ++k) {
    f32x4 o;
    o.x = (float)((res >> (31 - (4 * k + 0))) & 1u);
    o.y = (float)((res >> (31 - (4 * k + 1))) & 1u);
    o.z = (float)((res >> (31 - (4 * k + 2))) & 1u);
    o.w = (float)((res >> (31 - (4 * k + 3))) & 1u);
    __builtin_nontemporal_store(o, (f32x4*)(gout + 4 * k));  // 128B/value, aligned
  }
}

__global__ __launch_bounds__(THREADS)
void SpikeFP32Floor_43860206027309_kernel(const float* __restrict__ x,
                                          float* __restrict__ out, int nvals) {
  __shared__ uint32_t tile[VALS_PER_BLOCK * ROW_DW];  // 36,864 B

  const int lane     = threadIdx.x & 31;
  const int wave     = threadIdx.x >> 5;
  const int waveBase = blockIdx.x * VALS_PER_BLOCK + wave * 32;
  const int myVal    = waveBase + lane;

  if (waveBase + 32 <= nvals) {
    // ---- Stage this wave's 4KB (32 values x 128B) into LDS, coalesced ----
    uint32_t* wtile = &tile[(size_t)(wave * 32) * ROW_DW];
    const uint32_t* gsrc = (const uint32_t*)x + (size_t)waveBase * 32;

#if HAVE_ASYNC_LDS
    #pragma unroll
    for (int k = 0; k < 8; ++k) {
      const int idx = k * 32 + lane;      // 16B chunk id within wave tile
      const int v = idx >> 3;             // value row (8 chunks/value)
      const int c = idx & 7;              // chunk within value
      uint32_t* ldst = wtile + v * ROW_DW + c * 4;
      const uint32_t* g = gsrc + (size_t)idx * 4;
      __builtin_amdgcn_global_load_async_to_lds_b128(
          (gbl_i32x4*)(uintptr_t)g,       // global src (addrspace(1) int4*)
          (lds_i32x4*)(uintptr_t)ldst,    // LDS dst (addrspace(3) int4*; low
                                          // 32 bits of generic ptr = LDS addr)
          /*offset=*/0, /*cpol=*/0);
    }
    __builtin_amdgcn_s_wait_asynccnt(0);  // same-wave producer/consumer: no barrier
#else
    #pragma unroll
    for (int k = 0; k < 8; ++k) {
      const int idx = k * 32 + lane;
      const int v = idx >> 3, c = idx & 7;
      u32x4 d = *(const u32x4*)(gsrc + (size_t)idx * 4);  // global_load_b128
      *(u32x4*)(wtile + v * ROW_DW + c * 4) = d;          // ds_store_b128
    }
#endif

    // ---- Per-lane transposed read-back: 8 x ds_load_b128, pack, floor ----
    const uint32_t* row = &tile[(size_t)(wave * 32 + lane) * ROW_DW];
    uint32_t u = 0;
    #pragma unroll
    for (int k = 0; k < 8; ++k) {
      u32x4 q = *(const u32x4*)(row + 4 * k);
      // bit j (MSB-first) lives in dword j; 1.0f has bit23 set, 0.0f doesn't
      u |= ((q.x >> 23) & 1u) << (31 - (4 * k + 0));
      u |= ((q.y >> 23) & 1u) << (31 - (4 * k + 1));
      u |= ((q.z >> 23) & 1u) << (31 - (4 * k + 2));
      u |= ((q.w >> 23) & 1u) << (31 - (4 * k + 3));
    }
    unpack_store(floor_bits(u), out + (size_t)myVal * 32);
  } else if (myVal < nvals) {
    // Safe tail path (not taken for 2048*1024 values, but kept deterministic)
    const uint32_t* g = (const uint32_t*)x + (size_t)myVal * 32;
    uint32_t u = 0;
    #pragma unroll
    for (int j = 0; j < 32; ++j) u |= ((g[j] >> 23) & 1u) << (31 - j);
    unpack_store(floor_bits(u), out + (size_t)myVal * 32);
  }
}

extern "C" void kernel_launch(void* const* d_in, const int* in_sizes, int n_in,
                              void* d_out, int out_size, void* d_ws, size_t ws_size,
                              hipStream_t stream) {
  const float* x = (const float*)d_in[0];
  float* out = (float*)d_out;
  const int nvals = in_sizes[0] / 32;  // 2048*1024 fp32 values, 32 bit-floats each
  const int blocks = (nvals + VALS_PER_BLOCK - 1) / VALS_PER_BLOCK;
  SpikeFP32Floor_43860206027309_kernel<<<blocks, THREADS, 0, stream>>>(x, out, nvals);
}